// DualGCNDiscriminator_59425167508077
// MI455X (gfx1250) — compile-verified
//
#include <hip/hip_runtime.h>
#include <hip/hip_bf16.h>
#include <math.h>
#include <cstdint>

// ---------------------------------------------------------------------------
// DualGCNDiscriminator on MI455X (gfx1250, wave32)
//   - fp32 WMMA (V_WMMA_F32_16X16X4_F32) for all dense GEMMs (exact numerics)
//   - W staged into LDS via TDM tensor_load_to_lds (K=128 layer) and
//     GLOBAL_LOAD_ASYNC_TO_LDS_B128 (K=64 layer), sync fallback if builtins absent
//   - CSR built once (deterministic via per-bucket sort), reused by all 4 convs
//   - aggregation = coalesced gathers, L2-resident feature matrix
// ---------------------------------------------------------------------------

#define N_NODES 50000
#define N_EDGES 800000
#define HID     64
#define RRELU_SLOPE 0.22916666666666666f  // (1/8 + 1/3)/2

typedef __attribute__((ext_vector_type(2))) float v2f;
typedef __attribute__((ext_vector_type(8))) float v8f;
typedef unsigned int u32x4 __attribute__((ext_vector_type(4)));
typedef int          i32x4 __attribute__((ext_vector_type(4)));
typedef int          i32x8 __attribute__((ext_vector_type(8)));
typedef int          gv4i  __attribute__((vector_size(16)));  // matches builtin pointee

// ------------------------- gfx1250 async/TDM feature detection (device pass)
#if defined(__HIP_DEVICE_COMPILE__)
# if __has_builtin(__builtin_amdgcn_global_load_async_to_lds_b128)
#  define USE_ASYNC_LDS 1
# endif
# if __has_builtin(__builtin_amdgcn_tensor_load_to_lds) && __has_builtin(__builtin_amdgcn_s_wait_tensorcnt)
#  define USE_TDM 1
# endif
#endif

#ifdef USE_ASYNC_LDS
#define AS1 __attribute__((address_space(1)))
#define AS3 __attribute__((address_space(3)))
__device__ __forceinline__ void async_ld_b128(const float* g, float* l) {
    // AS3 pointer = low 32 bits of the flat LDS address (aperture mapping)
    __builtin_amdgcn_global_load_async_to_lds_b128(
        (AS1 gv4i*)(uintptr_t)g,
        (AS3 gv4i*)(uintptr_t)(uint32_t)(uintptr_t)l,
        0, 0);
}
# if __has_builtin(__builtin_amdgcn_s_wait_asynccnt)
#  define WAIT_ASYNCCNT() __builtin_amdgcn_s_wait_asynccnt(0)
# else
#  define WAIT_ASYNCCNT() asm volatile("s_wait_asynccnt 0" ::: "memory")
# endif
#endif

#ifdef USE_TDM
# if defined(__clang_major__) && (__clang_major__ >= 23)
#  define TDM_LOAD(g0, g1, g2, g3) \
      __builtin_amdgcn_tensor_load_to_lds((g0), (g1), (g2), (g3), (i32x8){0,0,0,0,0,0,0,0}, 0)
# else
#  define TDM_LOAD(g0, g1, g2, g3) \
      __builtin_amdgcn_tensor_load_to_lds((g0), (g1), (g2), (g3), 0)
# endif
#endif

// ---------------------------------------------------------------- utilities
__global__ void k_zero_i32(int* __restrict__ p, int n) {
    int i = blockIdx.x * blockDim.x + threadIdx.x;
    if (i < n) p[i] = 0;
}

__global__ void k_count_deg(const int* __restrict__ dst, int* __restrict__ deg, int e) {
    int i = blockIdx.x * blockDim.x + threadIdx.x;
    if (i < e) atomicAdd(&deg[dst[i]], 1);
}

__global__ void k_dinv(const int* __restrict__ deg, float* __restrict__ dinv, int n) {
    int i = blockIdx.x * blockDim.x + threadIdx.x;
    if (i < n) dinv[i] = rsqrtf((float)deg[i] + 1.0f);
}

// single-block exclusive scan of deg -> rowptr[0..n], cursor[0..n-1]
__global__ __launch_bounds__(1024)
void k_scan_excl(const int* __restrict__ deg, int* __restrict__ rowptr,
                 int* __restrict__ cursor, int n) {
    __shared__ int buf[1024];
    __shared__ int carry;
    if (threadIdx.x == 0) carry = 0;
    __syncthreads();
    for (int base = 0; base < n; base += 1024) {
        int i = base + (int)threadIdx.x;
        int v = (i < n) ? deg[i] : 0;
        buf[threadIdx.x] = v;
        __syncthreads();
        for (int off = 1; off < 1024; off <<= 1) {
            int t = (threadIdx.x >= (unsigned)off) ? buf[threadIdx.x - off] : 0;
            __syncthreads();
            buf[threadIdx.x] += t;
            __syncthreads();
        }
        int excl = buf[threadIdx.x] - v + carry;
        if (i < n) { rowptr[i] = excl; cursor[i] = excl; }
        __syncthreads();
        if (threadIdx.x == 1023) carry += buf[1023];
        __syncthreads();
    }
    if (threadIdx.x == 0) rowptr[n] = carry;
}

__global__ void k_fill_csr(const int* __restrict__ src, const int* __restrict__ dst,
                           int* __restrict__ cursor, int* __restrict__ csr_src, int e) {
    int i = blockIdx.x * blockDim.x + threadIdx.x;
    if (i < e) {
        int pos = atomicAdd(&cursor[dst[i]], 1);
        csr_src[pos] = src[i];
    }
}

// canonicalize each bucket so summation order is deterministic across runs
__global__ void k_sort_buckets(int* __restrict__ csr_src,
                               const int* __restrict__ rowptr, int n) {
    int i = blockIdx.x * blockDim.x + threadIdx.x;
    if (i >= n) return;
    int b = rowptr[i], e = rowptr[i + 1];
    for (int j = b + 1; j < e; ++j) {
        int key = csr_src[j];
        int k2 = j - 1;
        while (k2 >= b && csr_src[k2] > key) { csr_src[k2 + 1] = csr_src[k2]; --k2; }
        csr_src[k2 + 1] = key;
    }
}

// ----------------------------------------------------------- W staging to LDS
// K=128: single TDM tensor_load_to_lds issued by wave 0 (2D D#: tile 64 x K)
// K=64 : per-lane GLOBAL_LOAD_ASYNC_TO_LDS_B128
// fallback: plain float4 copy
template<int K>
__device__ __forceinline__ void stage_W(const float* __restrict__ W, float* Wl, int tid) {
#ifdef USE_TDM
    if constexpr (K == 128) {
        if (tid < 32) {  // one wave issues the DMA for the whole block
            const unsigned lds_base = (unsigned)(uint32_t)(uintptr_t)(void*)Wl;
            const unsigned long long ga = (unsigned long long)(uintptr_t)W;
            // D# group0: count=1 | lds_addr | global_addr[56:0] | type=2
            u32x4 g0 = { 1u, lds_base, (unsigned)ga,
                         (unsigned)((ga >> 32) & 0x01FFFFFFull) | 0x80000000u };
            // D# group1: data_size=4B, tensor_dim0=64 (stride 64), tensor_dim1=K,
            //            tile_dim0=64, tile_dim1=K (whole matrix as one tile)
            i32x8 g1 = { 0x00020000,               // workgroup_mask=0, data_size=2(4B)
                         (int)(64u << 16),         // tensor_dim0[15:0]=64 at bits[63:48]
                         (int)((unsigned)K << 16), // tensor_dim1[15:0]=K at bits[95:80]
                         (int)(64u << 16),         // tile_dim0=64 at bits[127:112]
                         K,                        // tile_dim1=K at bits[143:128]
                         64,                       // tensor_dim0_stride=64
                         0, 0 };
            i32x4 gz = { 0, 0, 0, 0 };
            TDM_LOAD(g0, g1, gz, gz);
            __builtin_amdgcn_s_wait_tensorcnt(0);
        }
        return;
    }
#endif
#ifdef USE_ASYNC_LDS
    for (int idx = tid; idx < K * 64 / 4; idx += 128)
        async_ld_b128(W + idx * 4, Wl + idx * 4);
    WAIT_ASYNCCNT();
#else
    for (int idx = tid; idx < K * 64 / 4; idx += 128)
        ((float4*)Wl)[idx] = ((const float4*)W)[idx];
#endif
}

// ---------------------------------------------------- fp32 WMMA GEMM  (M x K) @ (K x 64)
// block = 128 threads = 4 waves; each wave owns one 16x16 output tile (n0 = wave*16)
// grid.x = M/16 (M = 50000 is an exact multiple of 16)
//
// V_WMMA_F32_16X16X4_F32 operand layouts (ISA 7.12.2, 32-bit):
//   A 16x4 : lane = half*16 + m (m=0..15); a[j] = A[m][k + 2*half + j], j=0,1
//   B 4x16 : lane = half*16 + nn;          b[j] = B[k + 2*half + j][nn]
//   C/D    : vgpr r, lane half/nn:         D[r + 8*half][nn]
template<int K>
__global__ __launch_bounds__(128)
void k_gemm_wmma(const float* __restrict__ A, const float* __restrict__ W,
                 float* __restrict__ Out) {
    __shared__ float Wl[K * 64];
    const int tid = threadIdx.x;

    stage_W<K>(W, Wl, tid);
    __syncthreads();

    const int wave = tid >> 5;
    const int lane = tid & 31;
    const int half = lane >> 4;
    const int l16  = lane & 15;
    const int m0 = blockIdx.x * 16;
    const int n0 = wave * 16;

    const float* __restrict__ arow = A + (size_t)(m0 + l16) * K;

    v8f c = {};
#pragma unroll
    for (int k = 0; k < K; k += 4) {
        const int ka = k + 2 * half;
        v2f a, b;
        a.x = arow[ka];
        a.y = arow[ka + 1];
        b.x = Wl[ka * 64 + n0 + l16];
        b.y = Wl[(ka + 1) * 64 + n0 + l16];
        c = __builtin_amdgcn_wmma_f32_16x16x4_f32(
                /*neg_a=*/false, a, /*neg_b=*/false, b,
                /*c_mod=*/(short)0, c, /*reuse_a=*/false, /*reuse_b=*/false);
    }

#pragma unroll
    for (int r = 0; r < 8; ++r) {
        int m = m0 + r + 8 * half;
        Out[(size_t)m * 64 + n0 + l16] = c[r];
    }
}

// --------------------------------------------------- normalized neighbor aggregation
// out[i][f] = act( bias[f] + H[i][f]*dinv[i]^2 + sum_e H[csr_src[e]][f]*dinv[src]*dinv[i] )
// block = 64 threads (one feature each), grid = N nodes. act: 0 = rrelu, 1 = tanh
__global__ __launch_bounds__(64)
void k_aggregate(const float* __restrict__ H, const int* __restrict__ rowptr,
                 const int* __restrict__ csr_src, const float* __restrict__ dinv,
                 const float* __restrict__ bias, float* __restrict__ Out, int act) {
    __shared__ int   s_src[64];
    __shared__ float s_w[64];
    const int i = blockIdx.x;
    const int f = threadIdx.x;
    const float di = dinv[i];
    float acc = bias[f] + H[(size_t)i * 64 + f] * di * di;

    const int b = rowptr[i], e = rowptr[i + 1];
    for (int base = b; base < e; base += 64) {
        const int cnt = (e - base < 64) ? (e - base) : 64;
        __syncthreads();
        if (f < cnt) {
            int s = csr_src[base + f];
            s_src[f] = s;
            s_w[f]   = dinv[s] * di;
        }
        __syncthreads();
        for (int j = 0; j < cnt; ++j) {
            if (j + 1 < cnt)
                __builtin_prefetch(&H[(size_t)s_src[j + 1] * 64 + f], 0, 0);
            acc += H[(size_t)s_src[j] * 64 + f] * s_w[j];
        }
    }

    if (act == 0) acc = (acc >= 0.0f) ? acc : acc * RRELU_SLOPE;
    else          acc = tanhf(acc);
    Out[(size_t)i * 64 + f] = acc;
}

// ---------------------------------------------------------- final projection to 1
__global__ __launch_bounds__(64)
void k_final(const float* __restrict__ ZE, const float* __restrict__ XF,
             const float* __restrict__ Wout, const float* __restrict__ bout,
             float* __restrict__ out) {
    __shared__ float red[2];
    const int i = blockIdx.x;
    const int f = threadIdx.x;
    float v = ZE[(size_t)i * 64 + f] * XF[(size_t)i * 64 + f] * Wout[f];
#pragma unroll
    for (int off = 16; off > 0; off >>= 1) v += __shfl_down(v, off, 32);
    if ((f & 31) == 0) red[f >> 5] = v;
    __syncthreads();
    if (f == 0) out[i] = red[0] + red[1] + bout[0];
}

// ---------------------------------------------------------------------------
static inline size_t align256(size_t x) { return (x + 255) & ~(size_t)255; }

extern "C" void kernel_launch(void* const* d_in, const int* in_sizes, int n_in,
                              void* d_out, int out_size, void* d_ws, size_t ws_size,
                              hipStream_t stream) {
    (void)in_sizes; (void)n_in; (void)out_size; (void)ws_size;
    const int N = N_NODES, E = N_EDGES;

    const float* z     = (const float*)d_in[0];
    const float* x     = (const float*)d_in[1];
    const int*   eidx  = (const int*)  d_in[2];
    const float* W_e1  = (const float*)d_in[3];
    const float* b_e1  = (const float*)d_in[4];
    const float* W_e2  = (const float*)d_in[5];
    const float* b_e2  = (const float*)d_in[6];
    const float* W_f1  = (const float*)d_in[7];
    const float* b_f1  = (const float*)d_in[8];
    const float* W_f2  = (const float*)d_in[9];
    const float* b_f2  = (const float*)d_in[10];
    const float* W_out = (const float*)d_in[11];
    const float* b_out = (const float*)d_in[12];
    const int* src = eidx;
    const int* dst = eidx + E;

    char* ws = (char*)d_ws;
    size_t o = 0;
    int*   deg     = (int*)  (ws + o); o = align256(o + (size_t)N * 4);
    float* dinv    = (float*)(ws + o); o = align256(o + (size_t)N * 4);
    int*   rowptr  = (int*)  (ws + o); o = align256(o + (size_t)(N + 1) * 4);
    int*   cursor  = (int*)  (ws + o); o = align256(o + (size_t)N * 4);
    int*   csr_src = (int*)  (ws + o); o = align256(o + (size_t)E * 4);
    float* G       = (float*)(ws + o); o = align256(o + (size_t)N * HID * 4);
    float* A1      = (float*)(ws + o); o = align256(o + (size_t)N * HID * 4);
    float* ZE      = (float*)(ws + o); o = align256(o + (size_t)N * HID * 4);
    float* XF      = (float*)(ws + o); o = align256(o + (size_t)N * HID * 4);

    const int TB = 256;
    // ---- graph structure (built once per call, reused by all four convs)
    k_zero_i32    <<<(N + TB - 1) / TB, TB, 0, stream>>>(deg, N);
    k_count_deg   <<<(E + TB - 1) / TB, TB, 0, stream>>>(dst, deg, E);
    k_dinv        <<<(N + TB - 1) / TB, TB, 0, stream>>>(deg, dinv, N);
    k_scan_excl   <<<1, 1024, 0, stream>>>(deg, rowptr, cursor, N);
    k_fill_csr    <<<(E + TB - 1) / TB, TB, 0, stream>>>(src, dst, cursor, csr_src, E);
    k_sort_buckets<<<(N + TB - 1) / TB, TB, 0, stream>>>(csr_src, rowptr, N);

    const int GT = N / 16;  // 3125 tile-rows, exact

    // ---- z branch
    k_gemm_wmma<128><<<GT, 128, 0, stream>>>(z, W_e1, G);
    k_aggregate     <<<N, 64, 0, stream>>>(G, rowptr, csr_src, dinv, b_e1, A1, 0);
    k_gemm_wmma<64> <<<GT, 128, 0, stream>>>(A1, W_e2, G);
    k_aggregate     <<<N, 64, 0, stream>>>(G, rowptr, csr_src, dinv, b_e2, ZE, 1);

    // ---- x branch
    k_gemm_wmma<128><<<GT, 128, 0, stream>>>(x, W_f1, G);
    k_aggregate     <<<N, 64, 0, stream>>>(G, rowptr, csr_src, dinv, b_f1, A1, 0);
    k_gemm_wmma<64> <<<GT, 128, 0, stream>>>(A1, W_f2, G);
    k_aggregate     <<<N, 64, 0, stream>>>(G, rowptr, csr_src, dinv, b_f2, XF, 1);

    // ---- (ze * xf) @ W_out + b_out
    k_final<<<N, 64, 0, stream>>>(ZE, XF, W_out, b_out, (float*)d_out);
}